// GIN_20950850470283
// MI455X (gfx1250) — compile-verified
//
#include <hip/hip_runtime.h>

#define D 128

typedef float v2f __attribute__((ext_vector_type(2)));
typedef float v8f __attribute__((ext_vector_type(8)));

// ---------------------------------------------------------------------------
// Transpose W[o][k] (row-major, 128x128) into K-pair-packed layout:
//   Wt[(k>>1)*256 + o*2 + (k&1)] = W[o*128 + k]
// so a WMMA B-fragment lane load is one contiguous 8-byte (v2f) read.
// ---------------------------------------------------------------------------
__global__ void k_transpose_pack(const float* __restrict__ W, float* __restrict__ Wt) {
    int t = blockIdx.x * 256 + threadIdx.x;     // 0..16383
    int o = t >> 7;
    int k = t & 127;
    Wt[(k >> 1) * 256 + o * 2 + (k & 1)] = W[o * 128 + k];
}

// ---------------------------------------------------------------------------
// z = (1 + eps[l]) * h   (elementwise, float4)
// ---------------------------------------------------------------------------
__global__ void k_init(const float* __restrict__ h, float* __restrict__ z,
                       const float* __restrict__ eps, int l, int n4) {
    int i = blockIdx.x * 256 + threadIdx.x;
    if (i >= n4) return;
    float e = 1.0f + eps[l];
    float4 v = ((const float4*)h)[i];
    v.x *= e; v.y *= e; v.z *= e; v.w *= e;
    ((float4*)z)[i] = v;
}

// ---------------------------------------------------------------------------
// z[dst] += relu(h[src]) : one wave (32 lanes) per edge, float4 per lane.
// ---------------------------------------------------------------------------
__global__ void k_scatter(const float* __restrict__ h, const int* __restrict__ src,
                          const int* __restrict__ dst, float* __restrict__ z,
                          int E, int N) {
    int w = (blockIdx.x * 256 + threadIdx.x) >> 5;
    int lane = threadIdx.x & 31;
    if (w >= E) return;
    int s = src[w];
    int d = dst[w];
    // clamp (identity for valid indices; keeps atomics in-bounds regardless)
    s = s < 0 ? 0 : (s >= N ? N - 1 : s);
    d = d < 0 ? 0 : (d >= N ? N - 1 : d);
    float4 v = ((const float4*)(h + (size_t)s * D))[lane];
    v.x = fmaxf(v.x, 0.0f); v.y = fmaxf(v.y, 0.0f);
    v.z = fmaxf(v.z, 0.0f); v.w = fmaxf(v.w, 0.0f);
    float* zp = z + (size_t)d * D + lane * 4;
    atomicAdd(zp + 0, v.x);
    atomicAdd(zp + 1, v.y);
    atomicAdd(zp + 2, v.z);
    atomicAdd(zp + 3, v.w);
}

// ---------------------------------------------------------------------------
// out[N,128] = in[N,128] @ W^T + bias, fp32 WMMA 16x16x4, K=128 (32 WMMAs).
// Block = 256 threads = 8 waves; block owns 16 output rows, wave w owns
// columns [16w, 16w+16). A tile staged in LDS (stride 132 floats -> the
// b64 A-fragment reads are bank-conflict free). Safe to run in-place
// (out == in): A tile is fully staged before any global store.
//
// Fragment layouts per CDNA5 ISA 7.12.2:
//   A 16x4 : lanes 0-15 row M=lane, VGPR0/1 = K0/K1 ; lanes 16-31 = K2/K3
//   B 4x16 : lanes = col N, VGPR0 = K0 (K2 for hi half), VGPR1 = K1 (K3)
//   C 16x16: VGPR v -> M=v (lo half) / v+8 (hi half), lane&15 = col N
// ---------------------------------------------------------------------------
__global__ __launch_bounds__(256)
void k_gemm(const float* __restrict__ in, const float* __restrict__ Wt,
            const float* __restrict__ bias, float* __restrict__ out, int N) {
    __shared__ float As[16 * 132];
    const int rb = blockIdx.x * 16;
    const int t = threadIdx.x;

    // cooperative load of 16x128 A tile (512 float4, 2 per thread)
#pragma unroll
    for (int i = 0; i < 2; ++i) {
        int idx = t + 256 * i;          // 0..511
        int row = idx >> 5;             // 32 float4 per row
        int c4  = idx & 31;
        int r = rb + row;
        if (r >= N) r = N - 1;          // clamp keeps EXEC uniform
        float4 v = ((const float4*)(in + (size_t)r * D))[c4];
        *(float4*)&As[row * 132 + c4 * 4] = v;
    }
    __syncthreads();

    const int wave = t >> 5;            // 0..7 -> output column tile
    const int lane = t & 31;
    const int half = lane >> 4;         // 0: K0/K1, 1: K2/K3
    const int mrow = lane & 15;
    const int col  = wave * 16 + mrow;

    float bv = bias[col];
    v8f acc = { bv, bv, bv, bv, bv, bv, bv, bv };

#pragma unroll
    for (int k0 = 0; k0 < D; k0 += 4) {
        int kk = k0 + 2 * half;
        v2f a = *(const v2f*)&As[mrow * 132 + kk];
        v2f b = *(const v2f*)&Wt[(kk >> 1) * 256 + col * 2];
        acc = __builtin_amdgcn_wmma_f32_16x16x4_f32(
            /*neg_a=*/false, a, /*neg_b=*/false, b,
            /*c_mod=*/(short)0, acc, /*reuse_a=*/false, /*reuse_b=*/false);
    }

#pragma unroll
    for (int v = 0; v < 8; ++v) {
        int r = rb + v + 8 * half;
        if (r < N) out[(size_t)r * D + col] = acc[v];
    }
}

// ---------------------------------------------------------------------------
// BatchNorm stat buffers: stats[0:128]=sum, stats[128:256]=sumsq
// ---------------------------------------------------------------------------
__global__ void k_bn_zero(float* __restrict__ stats) {
    stats[threadIdx.x] = 0.0f;
}

__global__ __launch_bounds__(256)
void k_bn_reduce(const float* __restrict__ z, float* __restrict__ stats,
                 int N, int rpb) {
    __shared__ float sh[256];
    int col  = threadIdx.x & 127;
    int half = threadIdx.x >> 7;
    int r0 = blockIdx.x * rpb;
    int r1 = r0 + rpb; if (r1 > N) r1 = N;
    float s = 0.0f, q = 0.0f;
    for (int r = r0 + half; r < r1; r += 2) {
        float v = z[(size_t)r * D + col];
        s += v;
        q += v * v;
    }
    sh[threadIdx.x] = s;
    __syncthreads();
    if (half == 0) atomicAdd(&stats[col], s + sh[128 + col]);
    __syncthreads();
    sh[threadIdx.x] = q;
    __syncthreads();
    if (half == 0) atomicAdd(&stats[128 + col], q + sh[128 + col]);
}

// sc[0:128]=scale, sc[128:256]=shift
__global__ void k_bn_final(const float* __restrict__ stats,
                           const float* __restrict__ g, const float* __restrict__ b,
                           float* __restrict__ sc, float invN) {
    int c = threadIdx.x;                // 128 threads
    float mean = stats[c] * invN;
    float var  = stats[128 + c] * invN - mean * mean;   // biased variance
    float scale = g[c] * rsqrtf(var + 1e-5f);
    sc[c]       = scale;
    sc[128 + c] = b[c] - mean * scale;
}

__global__ void k_bn_apply(const float* __restrict__ in, float* __restrict__ out,
                           const float* __restrict__ sc, int n4, int do_relu) {
    int i = blockIdx.x * 256 + threadIdx.x;
    if (i >= n4) return;
    int c4 = (i & 31) * 4;              // (i*4) % 128
    float4 v = ((const float4*)in)[i];
    float4 s = *(const float4*)&sc[c4];
    float4 h = *(const float4*)&sc[128 + c4];
    v.x = v.x * s.x + h.x;
    v.y = v.y * s.y + h.y;
    v.z = v.z * s.z + h.z;
    v.w = v.w * s.w + h.w;
    if (do_relu) {
        v.x = fmaxf(v.x, 0.0f); v.y = fmaxf(v.y, 0.0f);
        v.z = fmaxf(v.z, 0.0f); v.w = fmaxf(v.w, 0.0f);
    }
    ((float4*)out)[i] = v;
}

// ---------------------------------------------------------------------------
extern "C" void kernel_launch(void* const* d_in, const int* in_sizes, int n_in,
                              void* d_out, int out_size, void* d_ws, size_t ws_size,
                              hipStream_t stream) {
    const float* x    = (const float*)d_in[0];
    const int*   ei   = (const int*)  d_in[1];
    const float* W1   = (const float*)d_in[2];
    const float* b1   = (const float*)d_in[3];
    const float* g1   = (const float*)d_in[4];
    const float* bt1  = (const float*)d_in[5];
    const float* W2   = (const float*)d_in[6];
    const float* b2   = (const float*)d_in[7];
    const float* eps  = (const float*)d_in[8];
    const float* gO   = (const float*)d_in[9];
    const float* btO  = (const float*)d_in[10];

    const int N = in_sizes[0] / D;
    const int E = in_sizes[1] / 2;
    const int L = in_sizes[2] / (D * D);

    const int* src = ei;        // edge_index row 0
    const int* dst = ei + E;    // edge_index row 1

    float* ws    = (float*)d_ws;
    float* bufA  = ws;
    float* bufB  = bufA + (size_t)N * D;
    float* Wt1   = bufB + (size_t)N * D;          // L * 128*128
    float* Wt2   = Wt1  + (size_t)L * D * D;
    float* stats = Wt2  + (size_t)L * D * D;      // 256
    float* sc    = stats + 256;                   // 256

    for (int l = 0; l < L; ++l) {
        k_transpose_pack<<<64, 256, 0, stream>>>(W1 + (size_t)l * D * D, Wt1 + (size_t)l * D * D);
        k_transpose_pack<<<64, 256, 0, stream>>>(W2 + (size_t)l * D * D, Wt2 + (size_t)l * D * D);
    }

    const int n4      = N * D / 4;
    const int nEltBlk = (n4 + 255) / 256;
    const int nEdgBlk = (E * 32 + 255) / 256;     // wave per edge
    const int nGemBlk = (N + 15) / 16;
    const int rpb     = 512;
    const int nRedBlk = (N + rpb - 1) / rpb;
    const float invN  = 1.0f / (float)N;

    const float* h = x;
    for (int l = 0; l < L; ++l) {
        float* z = (l & 1) ? bufB : bufA;
        const int last = (l == L - 1);

        // z = (1+eps)*h ; z[dst] += relu(h[src])
        k_init<<<nEltBlk, 256, 0, stream>>>(h, z, eps, l, n4);
        k_scatter<<<nEdgBlk, 256, 0, stream>>>(h, src, dst, z, E, N);

        // z = z @ W1^T + b1   (in-place WMMA GEMM)
        k_gemm<<<nGemBlk, 256, 0, stream>>>(z, Wt1 + (size_t)l * D * D, b1 + l * D, z, N);

        // inner BN + ReLU
        k_bn_zero<<<1, 256, 0, stream>>>(stats);
        k_bn_reduce<<<nRedBlk, 256, 0, stream>>>(z, stats, N, rpb);
        k_bn_final<<<1, 128, 0, stream>>>(stats, g1 + l * D, bt1 + l * D, sc, invN);
        k_bn_apply<<<nEltBlk, 256, 0, stream>>>(z, z, sc, n4, 1);

        // z = z @ W2^T + b2   (in-place WMMA GEMM)
        k_gemm<<<nGemBlk, 256, 0, stream>>>(z, Wt2 + (size_t)l * D * D, b2 + l * D, z, N);

        // outer BN (+ ReLU except last layer); last layer writes d_out
        k_bn_zero<<<1, 256, 0, stream>>>(stats);
        k_bn_reduce<<<nRedBlk, 256, 0, stream>>>(z, stats, N, rpb);
        k_bn_final<<<1, 128, 0, stream>>>(stats, gO + l * D, btO + l * D, sc, invN);
        float* outp = last ? (float*)d_out : z;
        k_bn_apply<<<nEltBlk, 256, 0, stream>>>(z, outp, sc, n4, last ? 0 : 1);

        h = z;
    }
}